// UserMemory_61873298866791
// MI455X (gfx1250) — compile-verified
//
#include <hip/hip_runtime.h>
#include <hip/hip_bf16.h>
#include <math.h>

typedef __attribute__((ext_vector_type(2))) float v2f;
typedef __attribute__((ext_vector_type(8))) float v8f;

#define NPROTO 20
#define DIM    64
#define EPSN   1e-12f

// Butterfly sum within each 16-lane half of a wave32 using ds_swizzle
// group-of-32 XOR mode (offset[14:10]=xor, [4:0]=and_mask 0x1F).
__device__ __forceinline__ float xor16_reduce(float v) {
  v += __int_as_float(__builtin_amdgcn_ds_swizzle(__float_as_int(v), 0x201F)); // xor 8
  v += __int_as_float(__builtin_amdgcn_ds_swizzle(__float_as_int(v), 0x101F)); // xor 4
  v += __int_as_float(__builtin_amdgcn_ds_swizzle(__float_as_int(v), 0x081F)); // xor 2
  v += __int_as_float(__builtin_amdgcn_ds_swizzle(__float_as_int(v), 0x041F)); // xor 1
  return v;
}

__global__ __launch_bounds__(128)
void usermem_fused_kernel(const int* __restrict__ uidx,
                          const float* __restrict__ x,
                          const float* __restrict__ proto,
                          const float* __restrict__ W,
                          const float* __restrict__ bias,
                          float* __restrict__ out,
                          int nrows) {
  __shared__ __align__(16) float hbuf[16][68]; // 16x64 h-tile, padded rows (272 B, 16B aligned)
  __shared__ float rnorm[16];

  const int tid  = threadIdx.x;
  const int wave = tid >> 5;       // 0..3
  const int lane = tid & 31;       // 0..31
  const int half = lane >> 4;      // 0..1
  const int ll   = lane & 15;      // 0..15
  const int row0 = blockIdx.x << 4;

  // Row/proto job mapping for phase 3: each 16-lane half owns rows {pair, pair+8}
  const int pair = (wave << 1) + half; // 0..7

  // Prefetch this block's anchor gather while the GEMM runs.
  {
    int r0 = row0 + pair;      if (r0 >= nrows) r0 = nrows - 1;
    int r1 = row0 + pair + 8;  if (r1 >= nrows) r1 = nrows - 1;
    const float* a0 = proto + (size_t)uidx[r0] * (NPROTO * DIM);
    const float* a1 = proto + (size_t)uidx[r1] * (NPROTO * DIM);
    __builtin_prefetch(a0 + ll * 80, 0, 3);  // cover 5120B block every 320B
    __builtin_prefetch(a0 + ll * 80 + 40, 0, 3);
    __builtin_prefetch(a1 + ll * 80, 0, 3);
    __builtin_prefetch(a1 + ll * 80 + 40, 0, 3);
  }

  // ---- Phase 1: h = x @ W^T + b via V_WMMA_F32_16X16X4_F32 ----
  // Wave w computes the 16x16 tile for output columns [16w, 16w+16).
  // A frag (16x4 f32): lane -> row M = ll, K = k4 + 2*half (+0,+1) -> float2
  // B frag (4x16 f32): lane -> col N = ncol, K = k4 + 2*half       -> float2
  const int ncol = (wave << 4) + ll;
  int arow = row0 + ll; if (arow >= nrows) arow = nrows - 1; // safe clamp (N%16==0 in harness)
  const float* xrow = x + (size_t)arow * DIM;
  const float* wrow = W + (size_t)ncol * DIM;

  v8f acc = {0.f, 0.f, 0.f, 0.f, 0.f, 0.f, 0.f, 0.f};
  #pragma unroll
  for (int k4 = 0; k4 < DIM; k4 += 4) {
    const int k = k4 + (half << 1);
    v2f a = *(const v2f*)(xrow + k);
    v2f b = *(const v2f*)(wrow + k);
    acc = __builtin_amdgcn_wmma_f32_16x16x4_f32(false, a, false, b,
                                                (short)0, acc, false, false);
  }

  // C/D layout: VGPR r, lanes 0-15 -> M=r, N=ll ; lanes 16-31 -> M=r+8, N=ll
  const float bv = bias[ncol];
  #pragma unroll
  for (int r = 0; r < 8; ++r) {
    const int m = r + (half << 3);
    hbuf[m][ncol] = acc[r] + bv;
  }
  __syncthreads();

  // ---- Phase 2: inverse row norms with F.normalize eps clamp ----
  if (tid < 16) {
    float s = 0.f;
    #pragma unroll
    for (int c = 0; c < DIM; ++c) { const float v = hbuf[tid][c]; s += v * v; }
    rnorm[tid] = 1.0f / fmaxf(sqrtf(s), EPSN);
  }
  __syncthreads();

  // ---- Phase 3: stream anchors, cosine-sum per row ----
  // Each 16-lane half: lane ll loads float4 -> half-wave covers one contiguous
  // 256B anchor row; 20 rows/user stream 5KB sequentially (coalesced).
  #pragma unroll
  for (int it = 0; it < 2; ++it) {
    const int m   = pair + (it << 3);
    const int row = row0 + m;
    if (row >= nrows) continue;
    const float  rn = rnorm[m];
    const float4 hv = *(const float4*)(&hbuf[m][ll << 2]);
    const float* ab = proto + (size_t)uidx[row] * (NPROTO * DIM) + (ll << 2);
    float score = 0.f;
    for (int p = 0; p < NPROTO; ++p) {
      const float4 av = *(const float4*)(ab + p * DIM);
      float d = hv.x * av.x + hv.y * av.y + hv.z * av.z + hv.w * av.w;
      float q = av.x * av.x + av.y * av.y + av.z * av.z + av.w * av.w;
      d = xor16_reduce(d);
      q = xor16_reduce(q);
      score += d * rn / fmaxf(sqrtf(q), EPSN);
    }
    if (ll == 0) out[row] = score;
  }
}

extern "C" void kernel_launch(void* const* d_in, const int* in_sizes, int n_in,
                              void* d_out, int out_size, void* d_ws, size_t ws_size,
                              hipStream_t stream) {
  const int*   uidx  = (const int*)d_in[0];
  const float* x     = (const float*)d_in[1];
  const float* proto = (const float*)d_in[2];
  const float* W     = (const float*)d_in[3];
  const float* b     = (const float*)d_in[4];
  float*       out   = (float*)d_out;

  const int n      = in_sizes[0];          // N = 16384
  const int blocks = (n + 15) / 16;        // one block per 16-row tile
  usermem_fused_kernel<<<blocks, 128, 0, stream>>>(uidx, x, proto, W, b, out, n);
}